// ChebyshevDistance_40475771797614
// MI455X (gfx1250) — compile-verified
//
#include <hip/hip_runtime.h>
#include <stdint.h>

#define TILE 64          // output tile is TILE x TILE per block
#define DDIM 32          // feature dimension
#define PAD  36          // floats per LDS row: 144B -> 16B aligned, bank-conflict free

__global__ __launch_bounds__(256)
void ChebyshevDistance_40475771797614_kernel(const float* __restrict__ A,
                                             const float* __restrict__ B,
                                             float* __restrict__ out,
                                             int M /* output cols & row stride */) {
    __shared__ float As[TILE * PAD];
    __shared__ float Bs[TILE * PAD];

    const int t  = threadIdx.x;        // 0..255
    const int bi = blockIdx.y;         // row-tile index (over A rows)
    const int bj = blockIdx.x;         // col-tile index (over B rows)

    // Uniform (SGPR) global bases for this block's tiles: 64 rows * 32 f32 = 8192 B each
    const float* gA = A + (size_t)bi * TILE * DDIM;
    const float* gB = B + (size_t)bj * TILE * DDIM;

    // LDS byte offsets of the tile buffers (low 32 bits of flat shared pointer
    // == wave-relative LDS offset, per CDNA5 aperture mapping).
    const uint32_t asBase = (uint32_t)(uintptr_t)(void*)As;
    const uint32_t bsBase = (uint32_t)(uintptr_t)(void*)Bs;

    // Stage 8 KB of A-rows and 8 KB of B-rows into LDS with async b128 copies.
    // 512 chunks of 16 B per matrix; 256 threads -> 2 chunks each.
    // Global layout is dense (32 f32/row); LDS layout is padded (36 f32/row).
#pragma unroll
    for (int k = 0; k < 2; ++k) {
        const int c   = t + k * 256;       // chunk id 0..511
        const int row = c >> 3;            // 8 chunks (128 B) per row
        const int cic = c & 7;
        const uint32_t ldsA = asBase + (uint32_t)(row * (PAD * 4) + cic * 16);
        const uint32_t ldsB = bsBase + (uint32_t)(row * (PAD * 4) + cic * 16);
        const uint32_t goff = (uint32_t)c * 16u;   // byte offset in dense 8 KB region
        asm volatile("global_load_async_to_lds_b128 %0, %1, %2"
                     :: "v"(ldsA), "v"(goff), "s"(gA) : "memory");
        asm volatile("global_load_async_to_lds_b128 %0, %1, %2"
                     :: "v"(ldsB), "v"(goff), "s"(gB) : "memory");
    }
    // Wait for this wave's async copies to land in LDS, then make them visible
    // workgroup-wide.
    asm volatile("s_wait_asynccnt 0" ::: "memory");
    __syncthreads();

    const int tx = t & 15;   // 16 threads across columns
    const int ty = t >> 4;   // 16 threads across rows

    float acc[4][4];
#pragma unroll
    for (int ii = 0; ii < 4; ++ii)
#pragma unroll
        for (int jj = 0; jj < 4; ++jj) acc[ii][jj] = 0.0f;

    // 8 d-blocks of 4 features; all LDS reads are 16B-aligned float4.
#pragma unroll
    for (int d0 = 0; d0 < DDIM; d0 += 4) {
        float4 a[4], b[4];
#pragma unroll
        for (int ii = 0; ii < 4; ++ii)
            a[ii] = *(const float4*)&As[(ty * 4 + ii) * PAD + d0];
#pragma unroll
        for (int jj = 0; jj < 4; ++jj)
            b[jj] = *(const float4*)&Bs[(tx * 4 + jj) * PAD + d0];

#pragma unroll
        for (int ii = 0; ii < 4; ++ii) {
#pragma unroll
            for (int jj = 0; jj < 4; ++jj) {
                float r  = acc[ii][jj];
                float e0 = a[ii].x - b[jj].x;
                float e1 = a[ii].y - b[jj].y;
                float e2 = a[ii].z - b[jj].z;
                float e3 = a[ii].w - b[jj].w;
                // Shaped so the backend folds each pair into
                // v_max3_num_f32 with |src| modifiers (3 VALU per 2 features).
                r = fmaxf(fmaxf(r, fabsf(e0)), fabsf(e1));
                r = fmaxf(fmaxf(r, fabsf(e2)), fabsf(e3));
                acc[ii][jj] = r;
            }
        }
    }

    // Coalesced b128 stores: each thread writes 4 rows x 4 consecutive floats.
    const int gj = bj * TILE + tx * 4;
#pragma unroll
    for (int ii = 0; ii < 4; ++ii) {
        const size_t gi = (size_t)(bi * TILE + ty * 4 + ii);
        float4 v = make_float4(acc[ii][0], acc[ii][1], acc[ii][2], acc[ii][3]);
        *(float4*)&out[gi * (size_t)M + gj] = v;
    }
}

extern "C" void kernel_launch(void* const* d_in, const int* in_sizes, int n_in,
                              void* d_out, int out_size, void* d_ws, size_t ws_size,
                              hipStream_t stream) {
    const float* A = (const float*)d_in[0];   // embed1: [N, 32] f32
    const float* B = (const float*)d_in[1];   // embed2: [M, 32] f32
    float* out = (float*)d_out;               // [N, M] f32

    const int N = in_sizes[0] / DDIM;
    const int M = in_sizes[1] / DDIM;

    dim3 grid((unsigned)(M / TILE), (unsigned)(N / TILE));
    ChebyshevDistance_40475771797614_kernel<<<grid, 256, 0, stream>>>(A, B, out, M);
}